// Quantizer_54185307407163
// MI455X (gfx1250) — compile-verified
//
#include <hip/hip_runtime.h>
#include <hip/hip_bf16.h>

typedef __attribute__((ext_vector_type(2))) float v2f;
typedef __attribute__((ext_vector_type(8))) float v8f;

#define K_CODES 4096
#define DIM     16
#define N_TILES (K_CODES / 16)     // 256 codebook tiles of 16 codes
#define N_X     65536              // 256*256 query vectors
#define EPSV    1e-12f

#define CHUNK   16                 // codebook tiles staged per LDS buffer
#define NCHUNK  (N_TILES / CHUNK)  // 16 chunks
#define MTILES  2                  // A row-tiles per wave (32 rows/wave)

// ---------------------------------------------------------------------------
// Prep: normalize codebook rows, compute ||en||^2, pre-pack normalized
// codebook into the WMMA B-operand lane layout (mirrors the documented 32-bit
// A 16x4 layout):
//   step s, v0: lanes 0-15 -> K=4s+0,N=lane ; lanes 16-31 -> K=4s+2,N=lane-16
//           v1: lanes 0-15 -> K=4s+1       ; lanes 16-31 -> K=4s+3
// Packed as bpack[(tile*32 + lane)*8 + (s*2 + v)]  (1 KB per tile, linear).
// ---------------------------------------------------------------------------
__global__ __launch_bounds__(256)
void vq_prep(const float* __restrict__ cb,
             float* __restrict__ bpack,
             float* __restrict__ en2,
             float* __restrict__ loss_slot) {
    const int t = blockIdx.x * (blockDim.x >> 5) + (threadIdx.x >> 5);
    const int l = threadIdx.x & 31;
    if (t == 0 && l == 0) *loss_slot = 0.0f;   // zero loss accumulator each call
    if (t >= N_TILES) return;

    const int  n  = t * 16 + (l & 15);
    const bool hi = (l >= 16);
    const float4* row = (const float4*)(cb + n * DIM);
    float4 q0 = row[0], q1 = row[1], q2 = row[2], q3 = row[3];

    float ss = q0.x*q0.x + q0.y*q0.y + q0.z*q0.z + q0.w*q0.w
             + q1.x*q1.x + q1.y*q1.y + q1.z*q1.z + q1.w*q1.w
             + q2.x*q2.x + q2.y*q2.y + q2.z*q2.z + q2.w*q2.w
             + q3.x*q3.x + q3.y*q3.y + q3.z*q3.z + q3.w*q3.w;
    float inv = 1.0f / fmaxf(sqrtf(ss), EPSV);

    float4 o01, o23;
    o01.x = (hi ? q0.z : q0.x) * inv;  o01.y = (hi ? q0.w : q0.y) * inv;
    o01.z = (hi ? q1.z : q1.x) * inv;  o01.w = (hi ? q1.w : q1.y) * inv;
    o23.x = (hi ? q2.z : q2.x) * inv;  o23.y = (hi ? q2.w : q2.y) * inv;
    o23.z = (hi ? q3.z : q3.x) * inv;  o23.w = (hi ? q3.w : q3.y) * inv;

    float4* dst = (float4*)(bpack + (size_t)(t * 32 + l) * 8);
    dst[0] = o01;
    dst[1] = o23;
    if (!hi) en2[n] = ss * inv * inv;          // ||en||^2 for exact d2 argmin
}

// ---------------------------------------------------------------------------
// Main: 8 waves/block, 2 A-tiles (32 rows) per wave. B chunks double-buffered
// through LDS so all 8 waves share one global read of the codebook stream.
// ---------------------------------------------------------------------------
__global__ __launch_bounds__(256)
void vq_main(const float* __restrict__ x,
             const float* __restrict__ cb,
             const float* __restrict__ bpack,
             const float* __restrict__ en2,
             float* __restrict__ outq,
             float* __restrict__ outidx,
             float* __restrict__ outloss) {
    __shared__ float Bbuf[2][CHUNK * 256];     // 2 x 16 KB staged B tiles
    __shared__ float E2buf[2][CHUNK * 16];     // 2 x 1 KB staged ||en||^2
    __shared__ float invn[256];                // per-row 1/||x|| (8 waves * 32)

    const int  tid     = threadIdx.x;
    const int  wave    = tid >> 5;
    const int  l       = tid & 31;
    const int  lane16  = l & 15;
    const bool hi      = (l >= 16);
    const int  rowBase = (blockIdx.x * 8 + wave) * (MTILES * 16);
    const int  off     = hi ? 2 : 0;

    // ---- Load + normalize the wave's two A tiles (fp32 A layout per step s:
    //      lanes 0-15 hold K=4s,4s+1; lanes 16-31 hold K=4s+2,4s+3).
    v2f a[MTILES][4];
#pragma unroll
    for (int mt = 0; mt < MTILES; ++mt) {
        const int    r  = rowBase + mt * 16 + lane16;
        const float* xr = x + (size_t)r * DIM;
        float ss = 0.0f;
#pragma unroll
        for (int s = 0; s < 4; ++s) {
            float2 p = *(const float2*)(xr + 4 * s + off);
            a[mt][s][0] = p.x;  a[mt][s][1] = p.y;
            ss += p.x * p.x + p.y * p.y;
        }
        ss += __shfl_xor(ss, 16, 32);          // combine the two half-rows
        const float inv = 1.0f / fmaxf(sqrtf(ss), EPSV);
        a[mt][0][0] *= inv; a[mt][0][1] *= inv;
        a[mt][1][0] *= inv; a[mt][1][1] *= inv;
        a[mt][2][0] *= inv; a[mt][2][1] *= inv;
        a[mt][3][0] *= inv; a[mt][3][1] *= inv;
        if (!hi) invn[wave * 32 + mt * 16 + lane16] = inv;
    }

    float bestd[MTILES][8];
    int   bestn[MTILES][8];
#pragma unroll
    for (int mt = 0; mt < MTILES; ++mt)
#pragma unroll
        for (int v = 0; v < 8; ++v) { bestd[mt][v] = __builtin_inff(); bestn[mt][v] = 0; }

    // ---- Chunked, double-buffered sweep over all 256 codebook tiles.
    float4 breg[4];
    float2 ereg;
    {   // prologue: stage chunk 0
        const float4* src = (const float4*)(bpack + (size_t)0 * CHUNK * 256);
#pragma unroll
        for (int j = 0; j < 4; ++j) breg[j] = src[tid + 256 * j];
        if (tid < 128) ereg = ((const float2*)(en2))[tid];
        float4* dst = (float4*)Bbuf[0];
#pragma unroll
        for (int j = 0; j < 4; ++j) dst[tid + 256 * j] = breg[j];
        if (tid < 128) ((float2*)E2buf[0])[tid] = ereg;
    }
    __syncthreads();

    for (int c = 0; c < NCHUNK; ++c) {
        const int cur = c & 1;

        // Issue next chunk's global loads before compute (latency overlap).
        if (c + 1 < NCHUNK) {
            const float4* src = (const float4*)(bpack + (size_t)(c + 1) * CHUNK * 256);
#pragma unroll
            for (int j = 0; j < 4; ++j) breg[j] = src[tid + 256 * j];
            if (tid < 128) ereg = ((const float2*)(en2 + (c + 1) * CHUNK * 16))[tid];
        }

        // Process CHUNK tiles from LDS: 8 WMMAs per tile (2 M-tiles x 4 K-steps).
        for (int tt = 0; tt < CHUNK; ++tt) {
            const float* bt  = &Bbuf[cur][(tt * 32 + l) * 8];
            float4 b01 = ((const float4*)bt)[0];
            float4 b23 = ((const float4*)bt)[1];
            float  e2  = E2buf[cur][tt * 16 + lane16];
            v2f b0 = {b01.x, b01.y}, b1 = {b01.z, b01.w};
            v2f b2 = {b23.x, b23.y}, b3 = {b23.z, b23.w};
            const int n = (c * CHUNK + tt) * 16 + lane16;

#pragma unroll
            for (int mt = 0; mt < MTILES; ++mt) {
                v8f cc = {};
                cc = __builtin_amdgcn_wmma_f32_16x16x4_f32(false, a[mt][0], false, b0, (short)0, cc, false, false);
                cc = __builtin_amdgcn_wmma_f32_16x16x4_f32(false, a[mt][1], false, b1, (short)0, cc, false, false);
                cc = __builtin_amdgcn_wmma_f32_16x16x4_f32(false, a[mt][2], false, b2, (short)0, cc, false, false);
                cc = __builtin_amdgcn_wmma_f32_16x16x4_f32(false, a[mt][3], false, b3, (short)0, cc, false, false);
#pragma unroll
                for (int v = 0; v < 8; ++v) {
                    float d      = fmaf(-2.0f, cc[v], e2);   // ||en||^2 - 2 xn.en
                    bool  better = d < bestd[mt][v];
                    bestd[mt][v] = better ? d : bestd[mt][v];
                    bestn[mt][v] = better ? n : bestn[mt][v];
                }
            }
        }

        // Commit next chunk into the other buffer.
        if (c + 1 < NCHUNK) {
            float4* dst = (float4*)Bbuf[1 - cur];
#pragma unroll
            for (int j = 0; j < 4; ++j) dst[tid + 256 * j] = breg[j];
            if (tid < 128) ((float2*)E2buf[1 - cur])[tid] = ereg;
            __syncthreads();
        }
    }

    // ---- Branchless argmin reduction across the 16 lanes sharing each M
    //      (lowest index wins ties, matching jnp.argmin).
#pragma unroll
    for (int m = 8; m >= 1; m >>= 1) {
#pragma unroll
        for (int mt = 0; mt < MTILES; ++mt)
#pragma unroll
            for (int v = 0; v < 8; ++v) {
                float od   = __shfl_xor(bestd[mt][v], m, 32);
                int   on   = __shfl_xor(bestn[mt][v], m, 32);
                int   take = (int)(od < bestd[mt][v]) |
                             ((int)(od == bestd[mt][v]) & (int)(on < bestn[mt][v]));
                bestd[mt][v] = take ? od : bestd[mt][v];
                bestn[mt][v] = take ? on : bestn[mt][v];
            }
    }

    // ---- Outputs: indices, raw codebook gather (quantized_st == quantized in
    //      forward), and loss = 1.25 * mean((q - xn)^2).
    float lsum = 0.0f;
    const int col = lane16;
#pragma unroll
    for (int mt = 0; mt < MTILES; ++mt) {
        const int rb = rowBase + mt * 16 + (hi ? 8 : 0);
        if (lane16 == 0) {
#pragma unroll
            for (int v = 0; v < 8; ++v) outidx[rb + v] = (float)bestn[mt][v];
        }
#pragma unroll
        for (int v = 0; v < 8; ++v) {
            const int rr   = rb + v;
            float     q    = cb[(size_t)bestn[mt][v] * DIM + col];
            float     invr = invn[wave * 32 + (rr - rowBase)];
            float     xe   = x[(size_t)rr * DIM + col] * invr;
            float     diff = q - xe;
            lsum += diff * diff;
            outq[(size_t)rr * DIM + col] = q;
        }
    }
#pragma unroll
    for (int m = 16; m >= 1; m >>= 1) lsum += __shfl_xor(lsum, m, 32);
    if (l == 0) atomicAdd(outloss, lsum * (1.25f / (float)(N_X * DIM)));
}

// ---------------------------------------------------------------------------
extern "C" void kernel_launch(void* const* d_in, const int* in_sizes, int n_in,
                              void* d_out, int out_size, void* d_ws, size_t ws_size,
                              hipStream_t stream) {
    const float* x  = (const float*)d_in[0];   // [256,256,16] fp32
    const float* cb = (const float*)d_in[1];   // [4096,16]   fp32

    float* out     = (float*)d_out;
    float* outq    = out;                       // 65536*16 quantized_st
    float* outidx  = out + (size_t)N_X * DIM;   // 65536 indices (as float)
    float* outloss = outidx + N_X;              // scalar loss

    float* bpack = (float*)d_ws;                // 256 tiles * 32 lanes * 8 floats
    float* en2   = bpack + (size_t)N_TILES * 32 * 8;

    vq_prep<<<N_TILES / 8, 256, 0, stream>>>(cb, bpack, en2, outloss);

    // 2048 waves, 32 rows each; 8 waves/block -> 256 blocks.
    vq_main<<<N_X / 256, 256, 0, stream>>>(x, cb, bpack, en2, outq, outidx, outloss);
}